// PoseGCN_16552803958948
// MI455X (gfx1250) — compile-verified
//
#include <hip/hip_runtime.h>
#include <hip/hip_bf16.h>

typedef float v2f __attribute__((ext_vector_type(2)));
typedef float v8f __attribute__((ext_vector_type(8)));

// ---- workspace layout (float offsets) ----
#define WS_FCWF 0            // 272 kchunks * 32 lanes * 2 = 17408 : fc_W B-fragments
#define WS_W2F  17408        // 16 kc * 4 hc * 32 * 2 = 4096      : W2 B-fragments
#define WS_W1F  21504        // 4 hc * 32 * 2 = 256               : W1 (K padded to 4) B-fragments
#define WS_A    21760        // 289                                : normalized adjacency
#define WS_END  22049

// ---- LDS layout (float offsets), per-sample stride 1092 (== 4 mod 64 -> conflict-free frags)
#define L_A   0              // 292
#define L_XA  292            // 16*68 = 1088   aggregated input, K padded to 4
#define L_H1  1380           // 16*1092 = 17472  h1 / h2 (ping)
#define L_HG  18852          // 16*1092 = 17472  aggregated h (also x staging early)
#define L_OP  36324          // 8*256 = 2048     FC partials
#define SMEM_DW 38372

// Pre-pack weights into WMMA B-fragment order + build A_hat.
__global__ void posegcn_prep(const int* __restrict__ ei, const float* __restrict__ W1,
                             const float* __restrict__ W2, const float* __restrict__ fcW,
                             float* __restrict__ ws) {
    int idx = blockIdx.x * 256 + threadIdx.x;
    if (idx < 17408) {                                   // fc_W: B[k,c] = fcW[c,k]
        int v = idx & 1, l = (idx >> 1) & 31, kc = idx >> 6;
        int k = kc * 4 + ((l & 16) ? 2 : 0) + v;
        ws[WS_FCWF + idx] = fcW[(l & 15) * 1088 + k];
    } else if (idx < 21504) {                            // W2: B[k,h] = W2[k,h]
        int i2 = idx - 17408;
        int v = i2 & 1, l = (i2 >> 1) & 31, f = i2 >> 6;
        int k = (f >> 2) * 4 + ((l & 16) ? 2 : 0) + v;
        int h = (f & 3) * 16 + (l & 15);
        ws[idx] = W2[k * 64 + h];
    } else if (idx < 21760) {                            // W1 padded K=3->4
        int i3 = idx - 21504;
        int v = i3 & 1, l = (i3 >> 1) & 31, hc = (i3 >> 6) & 3;
        int k = ((l & 16) ? 2 : 0) + v;
        int h = hc * 16 + (l & 15);
        ws[idx] = (k < 3) ? W1[k * 64 + h] : 0.f;
    } else if (idx < WS_END) {                           // A_hat[dst,src], self-loops + edges
        int t = idx - WS_A;
        int dr = t / 17, sc = t % 17;
        int deg[17];
        #pragma unroll
        for (int i = 0; i < 17; i++) deg[i] = 1;         // self loops
        for (int j = 0; j < 16; j++) { int d = ei[16 + j]; deg[d]++; }
        float dd = rsqrtf((float)deg[dr]) * rsqrtf((float)deg[sc]);
        float val = 0.f;
        for (int j = 0; j < 16; j++)
            if (ei[j] == sc && ei[16 + j] == dr) val += dd;
        if (dr == sc) val += dd;
        ws[WS_A + t] = val;
    }
}

__global__ void __launch_bounds__(256, 2)
posegcn_main(const float* __restrict__ x, const float* __restrict__ b1,
             const float* __restrict__ b2, const float* __restrict__ fcb,
             const float* __restrict__ ws, float* __restrict__ out) {
    extern __shared__ float smem[];
    float* sA  = smem + L_A;
    float* sXA = smem + L_XA;
    float* sH1 = smem + L_H1;
    float* sHG = smem + L_HG;
    float* sOP = smem + L_OP;

    const int tid = threadIdx.x;
    const int wave = tid >> 5, lane = tid & 31;
    const int mm = lane & 15;                // row of A-frag / col(N) of B-frag
    const int koffl = (lane >> 4) << 1;      // K offset for upper half-wave

    // ---- phase 1: stage A_hat + x tile (x staging aliases sHG) ----
    for (int i = tid; i < 289; i += 256) sA[i] = ws[WS_A + i];
    {
        const float* xsrc = x + (size_t)blockIdx.x * 816;   // 16 samples * 51
        for (int i = tid; i < 816; i += 256) sHG[i] = xsrc[i];
    }
    __syncthreads();

    // ---- phase 2: xa[m,n,f] = sum_s A[n,s] x[m,s,f]  (K padded to 4) ----
    {
        int m = tid >> 4, u = tid & 15;
        for (int slot = u; slot < 68; slot += 16) {
            int n = slot >> 2, f = slot & 3;
            float v = 0.f;
            if (f < 3) {
                #pragma unroll
                for (int s = 0; s < 17; s++) v += sA[n * 17 + s] * sHG[m * 51 + s * 3 + f];
            }
            sXA[m * 68 + slot] = v;
        }
    }
    __syncthreads();

    // ---- phase 3: layer 1 GEMM: h1 = relu(xa @ W1 + b1), 68 WMMA jobs over 8 waves ----
    for (int job = wave; job < 68; job += 8) {
        int n = job >> 2, hc = job & 3;
        v2f a = *(const v2f*)&sXA[mm * 68 + n * 4 + koffl];
        v2f b = *(const v2f*)&ws[WS_W1F + hc * 64 + lane * 2];
        v8f c = {};
        c = __builtin_amdgcn_wmma_f32_16x16x4_f32(false, a, false, b, (short)0, c, false, false);
        float bb = b1[hc * 16 + mm];
        #pragma unroll
        for (int j = 0; j < 8; j++) {
            int m = (lane >> 4) ? j + 8 : j;
            sH1[m * 1092 + n * 64 + hc * 16 + mm] = fmaxf(c[j] + bb, 0.f);
        }
    }
    __syncthreads();

    // ---- phase 4: joint aggregation hagg[m,n,:] = sum_s A[n,s] h1[m,s,:] (4-wide) ----
    {
        int m = tid >> 4, hq = tid & 15;
        float acc[17][4];
        #pragma unroll
        for (int n = 0; n < 17; n++) { acc[n][0] = acc[n][1] = acc[n][2] = acc[n][3] = 0.f; }
        #pragma unroll
        for (int s = 0; s < 17; s++) {
            float4 v = *(const float4*)&sH1[m * 1092 + s * 64 + hq * 4];
            #pragma unroll
            for (int n = 0; n < 17; n++) {
                float w = sA[n * 17 + s];
                acc[n][0] += w * v.x; acc[n][1] += w * v.y;
                acc[n][2] += w * v.z; acc[n][3] += w * v.w;
            }
        }
        #pragma unroll
        for (int n = 0; n < 17; n++) {
            float4 o = make_float4(acc[n][0], acc[n][1], acc[n][2], acc[n][3]);
            *(float4*)&sHG[m * 1092 + n * 64 + hq * 4] = o;
        }
    }
    __syncthreads();

    // ---- phase 5: layer 2 GEMM: h2 = relu(hagg @ W2 + b2), 68 jobs x 16 K-steps ----
    for (int job = wave; job < 68; job += 8) {
        int n = job >> 2, hc = job & 3;
        v8f c = {};
        #pragma unroll
        for (int kc = 0; kc < 16; kc++) {
            v2f a = *(const v2f*)&sHG[mm * 1092 + n * 64 + (kc << 2) + koffl];
            v2f b = *(const v2f*)&ws[WS_W2F + (((kc << 2) | hc) << 6) + lane * 2];
            c = __builtin_amdgcn_wmma_f32_16x16x4_f32(false, a, false, b, (short)0, c, false, false);
        }
        float bb = b2[hc * 16 + mm];
        #pragma unroll
        for (int j = 0; j < 8; j++) {
            int m = (lane >> 4) ? j + 8 : j;
            sH1[m * 1092 + n * 64 + hc * 16 + mm] = fmaxf(c[j] + bb, 0.f);
        }
    }
    __syncthreads();

    // ---- phase 6: FC (M=16,N=16,K=1088): K split across 8 waves, 34 chunks each ----
    {
        v8f c = {};
        int k0 = wave * 34;
        for (int kc = k0; kc < k0 + 34; kc++) {
            v2f a = *(const v2f*)&sH1[mm * 1092 + (kc << 2) + koffl];
            v2f b = *(const v2f*)&ws[WS_FCWF + (kc << 6) + lane * 2];
            c = __builtin_amdgcn_wmma_f32_16x16x4_f32(false, a, false, b, (short)0, c, false, false);
        }
        #pragma unroll
        for (int j = 0; j < 8; j++) {
            int m = (lane >> 4) ? j + 8 : j;
            sOP[wave * 256 + m * 16 + mm] = c[j];
        }
    }
    __syncthreads();

    // ---- phase 7: fixed-order cross-wave reduction + bias, coalesced store ----
    {
        float v = fcb[tid & 15];
        #pragma unroll
        for (int w = 0; w < 8; w++) v += sOP[w * 256 + tid];
        out[(size_t)blockIdx.x * 256 + tid] = v;
    }
}

extern "C" void kernel_launch(void* const* d_in, const int* in_sizes, int n_in,
                              void* d_out, int out_size, void* d_ws, size_t ws_size,
                              hipStream_t stream) {
    const float* x   = (const float*)d_in[0];
    const int*   ei  = (const int*)d_in[1];
    const float* W1  = (const float*)d_in[2];
    const float* b1  = (const float*)d_in[3];
    const float* W2  = (const float*)d_in[4];
    const float* b2  = (const float*)d_in[5];
    const float* fcW = (const float*)d_in[6];
    const float* fcb = (const float*)d_in[7];
    float* out = (float*)d_out;
    float* ws  = (float*)d_ws;

    int B = in_sizes[0] / 51;                 // 65536 samples (17*3 feats each)

    posegcn_prep<<<(WS_END + 255) / 256, 256, 0, stream>>>(ei, W1, W2, fcW, ws);

    size_t smem = (size_t)SMEM_DW * sizeof(float);   // ~150 KB -> 2 blocks/WGP
    (void)hipFuncSetAttribute((const void*)posegcn_main,
                              hipFuncAttributeMaxDynamicSharedMemorySize, (int)smem);
    posegcn_main<<<B / 16, 256, smem, stream>>>(x, b1, b2, fcb, ws, out);
}